// TorchLIFNeuronGroup_67370857005439
// MI455X (gfx1250) — compile-verified
//
#include <hip/hip_runtime.h>

// LIF neuron scan: 65536 neurons x 1000 timesteps.
// Memory-bound streaming kernel: 786 MB moved once, ~34 us floor @ 23.3 TB/s.
// One neuron per lane (wave32), membrane state in VGPRs, time loop unrolled
// by 20 with all loads batched ahead of the dependent compute for deep MLP
// (~40 outstanding b32 loads/thread -> ~10.5 MB in flight across 2048 waves).
// Non-temporal hints on all traffic (zero reuse; don't pollute 192 MB L2).

#define N_NEURONS 65536
#define T_STEPS   1000
#define UNROLL    20              // 1000 = 20 * 50
#define GROUPS    (T_STEPS / UNROLL)
#define BLOCK     256

__global__ __launch_bounds__(BLOCK) void lif_scan_kernel(
    const float* __restrict__ inI,
    const float* __restrict__ inN,
    float* __restrict__ out)
{
    const int neuron = blockIdx.x * BLOCK + threadIdx.x;

    const float decay = 1.0f / 20.0f;   // DT / TAU
    const float nstd  = 0.1f;           // NOISE_STD
    const float eta   = 0.1f;           // ETA
    const float vmax  = 2.0f;           // MAX_TH
    const float vmin  = 0.5f;           // MIN_TH

    float V   = 0.0f;                   // V_RESET
    float Vth = 1.0f;                   // V_TH0

    const float* pI = inI + neuron;
    const float* pN = inN + neuron;
    float*       pO = out + neuron;

    for (int g = 0; g < GROUPS; ++g) {
        const size_t base = (size_t)g * UNROLL * N_NEURONS;

        // ---- Batch all loads for this group (40 b32 NT loads in flight) ----
        float bi[UNROLL], bn[UNROLL];
#pragma unroll
        for (int u = 0; u < UNROLL; ++u) {
            const size_t off = base + (size_t)u * N_NEURONS;
            bi[u] = __builtin_nontemporal_load(pI + off);
            bn[u] = __builtin_nontemporal_load(pN + off);
        }

        // ---- Sequential state update (register-resident V, Vth) ----
#pragma unroll
        for (int u = 0; u < UNROLL; ++u) {
            // I_eff = I + noise_std * n
            float Ieff = bi[u] + nstd * bn[u];
            // leaky integration: V += decay * (I_eff - V)
            V = V + decay * (Ieff - V);
            // spike + reset
            const bool sp = (V >= Vth);
            const float spike = sp ? 1.0f : 0.0f;
            V = sp ? 0.0f : V;
            // adaptive threshold: bump on spike, clamp to [0.5, 2.0]
            float Vb = Vth + eta;
            Vb  = fminf(Vb, vmax);
            Vth = sp ? Vb : Vth;
            Vth = fmaxf(Vth, vmin);     // never binds (Vth >= 1), kept for fidelity

            __builtin_nontemporal_store(spike, pO + base + (size_t)u * N_NEURONS);
        }
    }
}

extern "C" void kernel_launch(void* const* d_in, const int* in_sizes, int n_in,
                              void* d_out, int out_size, void* d_ws, size_t ws_size,
                              hipStream_t stream) {
    (void)in_sizes; (void)n_in; (void)out_size; (void)d_ws; (void)ws_size;

    const float* inI = (const float*)d_in[0];  // input_current (T, N) fp32
    const float* inN = (const float*)d_in[1];  // noise         (T, N) fp32
    float*       out = (float*)d_out;          // spikes        (T, N) fp32

    dim3 block(BLOCK);
    dim3 grid(N_NEURONS / BLOCK);              // 256 blocks = 2048 wave32s
    lif_scan_kernel<<<grid, block, 0, stream>>>(inI, inN, out);
}